// GATModel_64716567216322
// MI455X (gfx1250) — compile-verified
//
#include <hip/hip_runtime.h>
#include <hip/hip_bf16.h>
#include <math.h>

// ---------------- types ----------------
typedef float v2f __attribute__((ext_vector_type(2)));
typedef float v8f __attribute__((ext_vector_type(8)));

#define HID 64
#define HEADS 4
#define DH 16

// monotone float<->uint order mapping for atomic max on floats
__device__ __forceinline__ unsigned f2ord(float f) {
    unsigned b = __float_as_uint(f);
    return (b & 0x80000000u) ? ~b : (b | 0x80000000u);
}
__device__ __forceinline__ float ord2f(unsigned k) {
    unsigned b = (k & 0x80000000u) ? (k ^ 0x80000000u) : ~k;
    return __uint_as_float(b);
}

// ---------------------------------------------------------------
// GEMM: C[M,64] = A[M,K] @ W[K,64] + bias, fp32, via V_WMMA_F32_16X16X4_F32
// One wave computes a 16x64 tile. Weights staged in LDS with K-pair
// interleaving so each B fragment is ONE aligned ds_load_b64:
//   Wl[(k>>1)*128 + n*2 + (k&1)] = W[k*64 + n]
// Requires M % 16 == 0, K % 4 == 0.
// ---------------------------------------------------------------
__global__ __launch_bounds__(256) void gemm_n64(
    const float* __restrict__ A, const float* __restrict__ W,
    const float* __restrict__ bias, float* __restrict__ C, int M, int K) {
    __shared__ float Wl[64 * 64];
    const int tid = threadIdx.x;
    for (int i = tid; i < K * 64; i += blockDim.x) {
        const int k = i >> 6, n = i & 63;
        Wl[(k >> 1) * 128 + n * 2 + (k & 1)] = W[i];
    }
    __syncthreads();

    const int wave = tid >> 5;
    const int lane = tid & 31;
    const int half = lane >> 4;   // 0: lanes 0-15, 1: lanes 16-31
    const int l16  = lane & 15;
    const int m0 = (blockIdx.x * 8 + wave) * 16;
    if (m0 >= M) return;

    v8f acc0 = {}, acc1 = {}, acc2 = {}, acc3 = {};
    const float* arow = A + (size_t)(m0 + l16) * K;
    const v2f* Wp = (const v2f*)Wl;   // 64 v2f per K-pair row

    for (int kk = 0; kk < K; kk += 4) {
        // A 16x4 fragment: lanes 0-15 hold K=kk,kk+1 ; lanes 16-31 hold K=kk+2,kk+3
        v2f a;
        a.x = arow[kk + 2 * half + 0];
        a.y = arow[kk + 2 * half + 1];
        // B 4x16 fragments: pair row p = kk/2 + half, one b64 LDS load each
        const v2f* brow = Wp + ((kk >> 1) + half) * 64 + l16;
        acc0 = __builtin_amdgcn_wmma_f32_16x16x4_f32(false, a, false, brow[0],  (short)0, acc0, false, false);
        acc1 = __builtin_amdgcn_wmma_f32_16x16x4_f32(false, a, false, brow[16], (short)0, acc1, false, false);
        acc2 = __builtin_amdgcn_wmma_f32_16x16x4_f32(false, a, false, brow[32], (short)0, acc2, false, false);
        acc3 = __builtin_amdgcn_wmma_f32_16x16x4_f32(false, a, false, brow[48], (short)0, acc3, false, false);
    }
    // C/D layout: VGPR r -> rows m0+r (lanes 0-15) / m0+r+8 (lanes 16-31)
    #pragma unroll
    for (int r = 0; r < 8; ++r) {
        const int row = m0 + r + 8 * half;
        float* crow = C + (size_t)row * 64;
        crow[l16 +  0] = acc0[r] + bias[l16 +  0];
        crow[l16 + 16] = acc1[r] + bias[l16 + 16];
        crow[l16 + 32] = acc2[r] + bias[l16 + 32];
        crow[l16 + 48] = acc3[r] + bias[l16 + 48];
    }
}

// ---------------------------------------------------------------
// zero fill
// ---------------------------------------------------------------
__global__ void zero_u32(unsigned* __restrict__ p, long n) {
    long i = (long)blockIdx.x * blockDim.x + threadIdx.x;
    if (i < n) p[i] = 0u;
}

// ---------------------------------------------------------------
// Pass A: alpha[e,h] = dot(q[dst], k[src] + eproj) * 0.25  (eproj recomputed,
//         We/be staged in LDS) + ordered-uint atomic max into amax_u[dst,h]
// thread per (edge, head)
// ---------------------------------------------------------------
__global__ __launch_bounds__(256) void edge_alpha(
    const float* __restrict__ q, const float* __restrict__ k,
    const float* __restrict__ edge_attr,
    const float* __restrict__ We, const float* __restrict__ be,
    const int* __restrict__ src, const int* __restrict__ dst,
    float* __restrict__ alpha, unsigned* __restrict__ amax_u, int E) {
    __shared__ float sWe[16 * HID];
    __shared__ float sbe[HID];
    for (int i = threadIdx.x; i < 16 * HID; i += blockDim.x) sWe[i] = We[i];
    if (threadIdx.x < HID) sbe[threadIdx.x] = be[threadIdx.x];
    __syncthreads();

    long idx = (long)blockIdx.x * blockDim.x + threadIdx.x;
    if (idx >= (long)E * HEADS) return;
    const int e = (int)(idx >> 2), h = (int)(idx & 3);
    const int s = src[e], d = dst[e];
    const float* qrow = q + (size_t)d * HID + h * DH;
    const float* krow = k + (size_t)s * HID + h * DH;
    float ea[16];
    #pragma unroll
    for (int i = 0; i < 16; ++i) ea[i] = edge_attr[(size_t)e * 16 + i];
    float acc = 0.f;
    #pragma unroll
    for (int j = 0; j < DH; ++j) {
        float ep = sbe[h * DH + j];
        #pragma unroll
        for (int i = 0; i < 16; ++i) ep = fmaf(ea[i], sWe[i * HID + h * DH + j], ep);
        acc = fmaf(qrow[j], krow[j] + ep, acc);
    }
    acc *= 0.25f; // 1/sqrt(DH)
    alpha[idx] = acc;
    atomicMax(&amax_u[(size_t)d * HEADS + h], f2ord(acc));
}

// ---------------------------------------------------------------
// Pass B: ex = exp(alpha - amax[dst]); alpha <- ex; denom[dst,h] += ex
// ---------------------------------------------------------------
__global__ __launch_bounds__(256) void edge_exp(
    const int* __restrict__ dst, const unsigned* __restrict__ amax_u,
    float* __restrict__ alpha, float* __restrict__ denom, int E) {
    long idx = (long)blockIdx.x * blockDim.x + threadIdx.x;
    if (idx >= (long)E * HEADS) return;
    const int e = (int)(idx >> 2), h = (int)(idx & 3);
    const int d = dst[e];
    const float amax = ord2f(amax_u[(size_t)d * HEADS + h]);
    const float ex = expf(alpha[idx] - amax);
    alpha[idx] = ex;
    atomicAdd(&denom[(size_t)d * HEADS + h], ex);
}

// ---------------------------------------------------------------
// Pass C: out[dst,h,:] += (ex/denom) * (v[src,h,:] + eproj)  (atomic scatter)
// out is pre-initialized with the skip projection h@Ws+bs. We/be in LDS.
// ---------------------------------------------------------------
__global__ __launch_bounds__(256) void edge_scatter(
    const float* __restrict__ v, const float* __restrict__ edge_attr,
    const float* __restrict__ We, const float* __restrict__ be,
    const int* __restrict__ src, const int* __restrict__ dst,
    const float* __restrict__ alpha, const float* __restrict__ denom,
    float* __restrict__ out, int E) {
    __shared__ float sWe[16 * HID];
    __shared__ float sbe[HID];
    for (int i = threadIdx.x; i < 16 * HID; i += blockDim.x) sWe[i] = We[i];
    if (threadIdx.x < HID) sbe[threadIdx.x] = be[threadIdx.x];
    __syncthreads();

    long idx = (long)blockIdx.x * blockDim.x + threadIdx.x;
    if (idx >= (long)E * HEADS) return;
    const int e = (int)(idx >> 2), h = (int)(idx & 3);
    const int s = src[e], d = dst[e];
    const float w = alpha[idx] / (denom[(size_t)d * HEADS + h] + 1e-16f);
    float ea[16];
    #pragma unroll
    for (int i = 0; i < 16; ++i) ea[i] = edge_attr[(size_t)e * 16 + i];
    const float* vrow = v + (size_t)s * HID + h * DH;
    float* orow = out + (size_t)d * HID + h * DH;
    #pragma unroll
    for (int j = 0; j < DH; ++j) {
        float ep = sbe[h * DH + j];
        #pragma unroll
        for (int i = 0; i < 16; ++i) ep = fmaf(ea[i], sWe[i * HID + h * DH + j], ep);
        atomicAdd(&orow[j], w * (vrow[j] + ep));
    }
}

// ---------------------------------------------------------------
// BN reduction: sums[0..63]=sum per channel, sums[64..127]=sumsq
// ---------------------------------------------------------------
__global__ __launch_bounds__(256) void bn_reduce(
    const float* __restrict__ x, float* __restrict__ sums, int M) {
    __shared__ float ssum[64], ssq[64];
    const int c = threadIdx.x & 63;
    const int slot = threadIdx.x >> 6;           // 4 row-slots per block
    float lsum = 0.f, lsq = 0.f;
    for (int r = blockIdx.x * 4 + slot; r < M; r += gridDim.x * 4) {
        float vv = x[(size_t)r * 64 + c];
        lsum += vv; lsq += vv * vv;
    }
    if (threadIdx.x < 64) { ssum[c] = 0.f; ssq[c] = 0.f; }
    __syncthreads();
    atomicAdd(&ssum[c], lsum);
    atomicAdd(&ssq[c], lsq);
    __syncthreads();
    if (threadIdx.x < 64) {
        atomicAdd(&sums[c], ssum[c]);
        atomicAdd(&sums[64 + c], ssq[c]);
    }
}

// ---------------------------------------------------------------
// BN apply + exact GELU + residual:  out = gelu(gamma*(x-mu)*rsqrt(var+eps)+beta) + hin
// ---------------------------------------------------------------
__global__ __launch_bounds__(256) void bn_apply(
    const float* __restrict__ x, const float* __restrict__ hin,
    const float* __restrict__ sums, const float* __restrict__ gamma,
    const float* __restrict__ beta, float* __restrict__ out, int M) {
    long idx = (long)blockIdx.x * blockDim.x + threadIdx.x;
    if (idx >= (long)M * 64) return;
    const int c = (int)(idx & 63);
    const float inv = 1.0f / (float)M;
    const float mu = sums[c] * inv;
    const float var = sums[64 + c] * inv - mu * mu;
    const float xn = (x[idx] - mu) * rsqrtf(var + 1e-5f);
    const float y = gamma[c] * xn + beta[c];
    const float g = 0.5f * y * (1.0f + erff(y * 0.70710678118654752440f));
    out[idx] = g + hin[idx];
}

// ---------------------------------------------------------------
// Graph pooling: pool[g,c] += h[n,c]; cnt[g] += 1
// ---------------------------------------------------------------
__global__ __launch_bounds__(256) void pool_kernel(
    const float* __restrict__ h, const int* __restrict__ batch,
    float* __restrict__ pool, float* __restrict__ cnt, int Nn) {
    long idx = (long)blockIdx.x * blockDim.x + threadIdx.x;
    if (idx >= (long)Nn * 64) return;
    const int n = (int)(idx >> 6), c = (int)(idx & 63);
    const int g = batch[n];
    atomicAdd(&pool[(size_t)g * 64 + c], h[idx]);
    if (c == 0) atomicAdd(&cnt[g], 1.0f);
}

// ---------------------------------------------------------------
// Regressor: g=[mean,sum] (128) -> relu(g@rw1+rb1) (64) -> @rw2+rb2 (1)
// one block (64 threads) per graph
// ---------------------------------------------------------------
__global__ __launch_bounds__(64) void regressor(
    const float* __restrict__ pool, const float* __restrict__ cnt,
    const float* __restrict__ rw1, const float* __restrict__ rb1,
    const float* __restrict__ rw2, const float* __restrict__ rb2,
    float* __restrict__ out) {
    const int g = blockIdx.x, j = threadIdx.x;
    __shared__ float red[64];
    const float c = fmaxf(cnt[g], 1.0f);
    const float* s = pool + (size_t)g * 64;
    float acc = rb1[j];
    for (int i = 0; i < 64; ++i) acc = fmaf(s[i] / c, rw1[i * 64 + j], acc);        // mean half
    for (int i = 0; i < 64; ++i) acc = fmaf(s[i],     rw1[(64 + i) * 64 + j], acc); // sum half
    acc = fmaxf(acc, 0.f);
    red[j] = acc * rw2[j];
    __syncthreads();
    for (int off = 32; off > 0; off >>= 1) {
        if (j < off) red[j] += red[j + off];
        __syncthreads();
    }
    if (j == 0) out[g] = red[0] + rb2[0];
}

// ---------------------------------------------------------------
extern "C" void kernel_launch(void* const* d_in, const int* in_sizes, int n_in,
                              void* d_out, int out_size, void* d_ws, size_t ws_size,
                              hipStream_t stream) {
    const float* x         = (const float*)d_in[0];
    const int*   edge_idx  = (const int*)d_in[1];
    const float* edge_attr = (const float*)d_in[2];
    const int*   batch     = (const int*)d_in[3];
    const float* node_w    = (const float*)d_in[4];
    const float* node_b    = (const float*)d_in[5];
    const float* Wq = (const float*)d_in[6];   const float* bq = (const float*)d_in[7];
    const float* Wk = (const float*)d_in[8];   const float* bk = (const float*)d_in[9];
    const float* Wv = (const float*)d_in[10];  const float* bv = (const float*)d_in[11];
    const float* We = (const float*)d_in[12];  const float* be = (const float*)d_in[13];
    const float* Ws = (const float*)d_in[14];  const float* bs = (const float*)d_in[15];
    const float* gamma = (const float*)d_in[16];
    const float* beta  = (const float*)d_in[17];
    const float* rw1 = (const float*)d_in[18]; const float* rb1 = (const float*)d_in[19];
    const float* rw2 = (const float*)d_in[20]; const float* rb2 = (const float*)d_in[21];
    float* out = (float*)d_out;

    const int N = in_sizes[0] / 32;      // 100000
    const int E = in_sizes[1] / 2;       // 1000000
    const int G = out_size;              // 512
    const int L = in_sizes[7] / 64;      // 3
    const int* src = edge_idx;
    const int* dst = edge_idx + E;

    // ---- workspace carve-up (floats) ----
    float* ws = (float*)d_ws;
    const size_t nh = (size_t)N * HID;
    float*    hA     = ws;                 ws += nh;
    float*    hB     = ws;                 ws += nh;
    float*    qb     = ws;                 ws += nh;
    float*    kb     = ws;                 ws += nh;
    float*    vb     = ws;                 ws += nh;
    float*    attn   = ws;                 ws += nh;   // skip-proj + scatter target
    float*    alpha  = ws;                 ws += (size_t)E * HEADS;
    unsigned* amax_u = (unsigned*)ws;      ws += (size_t)N * HEADS;
    float*    denom  = ws;                 ws += (size_t)N * HEADS;
    float*    sums   = ws;                 ws += 128;
    float*    pool   = ws;                 ws += (size_t)G * 64;
    float*    cnt    = ws;                 ws += G;
    (void)ws_size;

    const int TB = 256;
    const int gemm_grid = (N + 127) / 128;
    const long eh = (long)E * HEADS;
    const int edge_grid = (int)((eh + TB - 1) / TB);
    const int node_grid = (int)(((long)N * 64 + TB - 1) / TB);

    // h = x @ node_w + node_b
    gemm_n64<<<gemm_grid, TB, 0, stream>>>(x, node_w, node_b, hA, N, 32);

    float* hin  = hA;
    float* hout = hB;
    for (int l = 0; l < L; ++l) {
        const float* Wq_l = Wq + (size_t)l * HID * HID; const float* bq_l = bq + l * HID;
        const float* Wk_l = Wk + (size_t)l * HID * HID; const float* bk_l = bk + l * HID;
        const float* Wv_l = Wv + (size_t)l * HID * HID; const float* bv_l = bv + l * HID;
        const float* We_l = We + (size_t)l * 16 * HID;  const float* be_l = be + l * HID;
        const float* Ws_l = Ws + (size_t)l * HID * HID; const float* bs_l = bs + l * HID;

        // clear per-layer scatter state
        {
            long nz = (long)N * HEADS;
            zero_u32<<<(int)((nz + TB - 1) / TB), TB, 0, stream>>>(amax_u, nz);
            zero_u32<<<(int)((nz + TB - 1) / TB), TB, 0, stream>>>((unsigned*)denom, nz);
            zero_u32<<<1, 128, 0, stream>>>((unsigned*)sums, 128);
        }

        // projections (WMMA)
        gemm_n64<<<gemm_grid, TB, 0, stream>>>(hin, Wq_l, bq_l, qb,   N, HID);
        gemm_n64<<<gemm_grid, TB, 0, stream>>>(hin, Wk_l, bk_l, kb,   N, HID);
        gemm_n64<<<gemm_grid, TB, 0, stream>>>(hin, Wv_l, bv_l, vb,   N, HID);
        gemm_n64<<<gemm_grid, TB, 0, stream>>>(hin, Ws_l, bs_l, attn, N, HID); // skip into attn

        // attention: scatter softmax + weighted aggregation
        edge_alpha  <<<edge_grid, TB, 0, stream>>>(qb, kb, edge_attr, We_l, be_l, src, dst, alpha, amax_u, E);
        edge_exp    <<<edge_grid, TB, 0, stream>>>(dst, amax_u, alpha, denom, E);
        edge_scatter<<<edge_grid, TB, 0, stream>>>(vb, edge_attr, We_l, be_l, src, dst, alpha, denom, attn, E);

        // BN + GELU + residual
        bn_reduce<<<1024, TB, 0, stream>>>(attn, sums, N);
        bn_apply <<<node_grid, TB, 0, stream>>>(attn, hin, sums, gamma + l * HID, beta + l * HID, hout, N);

        float* t = hin; hin = hout; hout = t;
    }

    // pooling + regressor
    {
        long nz = (long)G * 64 + G;
        zero_u32<<<(int)((nz + TB - 1) / TB), TB, 0, stream>>>((unsigned*)pool, nz); // pool then cnt (contiguous)
        pool_kernel<<<node_grid, TB, 0, stream>>>(hin, batch, pool, cnt, N);
        regressor<<<G, 64, 0, stream>>>(pool, cnt, rw1, rb1, rw2, rb2, out);
    }
}